// channel_crossmodel_56556129353812
// MI455X (gfx1250) — compile-verified
//
#include <hip/hip_runtime.h>
#include <math.h>

// Problem constants (from reference)
#define Bb   16
#define Cc   512
#define Ll   1024
#define KN   8
#define KS   15
#define SPSTR 1040          // padded-s row stride (>= L + K - 1 = 1038)
#define MSTR 33             // 32x32 matrix row stride (bank-conflict-free, gcd(33,64)=1)
#define MSZ  (32*MSTR)      // 1056 floats per matrix plane

typedef float v2f __attribute__((ext_vector_type(2)));
typedef float v8f __attribute__((ext_vector_type(8)));

// ---------------------------------------------------------------------------
// FP32 WMMA 16x16x4:  D = A(16x4) * B(4x16) + C   (wave32, CDNA5)
// ---------------------------------------------------------------------------
__device__ __forceinline__ v8f wmma4(v2f a, v2f b, v8f c) {
  return __builtin_amdgcn_wmma_f32_16x16x4_f32(false, a, false, b, (short)0, c,
                                               false, false);
}

// ---------------------------------------------------------------------------
// Complex 32x32x32 matmul in LDS, one wave:  D = A * B
// sA/sB = +-1 applied to the imaginary planes (conjugation for inverse FFT).
// Layouts per CDNA5 ISA 7.12.2:
//   A 16x4 f32 : lanes 0-15 hold M, v0=K{0|2}, v1=K{1|3} (lo|hi half-wave)
//   B 4x16 f32 : v0 = rows K{0|2}, v1 = rows K{1|3}, N striped on lanes&15
//   C/D 16x16  : VGPR r -> M = r (lanes 0-15) / r+8 (lanes 16-31)
// ---------------------------------------------------------------------------
__device__ __forceinline__ void cmatmul32(
    float* __restrict__ Dr, float* __restrict__ Di,
    const float* __restrict__ Ar, const float* __restrict__ Ai, float sA,
    const float* __restrict__ Br, const float* __restrict__ Bi, float sB,
    int lane) {
  const int ml   = lane & 15;
  const int hi   = lane >> 4;         // 0 or 1
  const int kofs = hi ? 2 : 0;
  for (int tm = 0; tm < 2; ++tm) {
    const int arow = tm * 16 + ml;
    for (int tn = 0; tn < 2; ++tn) {
      const int bcol = tn * 16 + ml;
      v8f accR = {};
      v8f accI = {};
#pragma unroll
      for (int kb = 0; kb < 8; ++kb) {
        const int k0 = kb * 4 + kofs;
        v2f aR, aI, bR, bI;
        aR.x = Ar[arow * MSTR + k0];
        aR.y = Ar[arow * MSTR + k0 + 1];
        aI.x = sA * Ai[arow * MSTR + k0];
        aI.y = sA * Ai[arow * MSTR + k0 + 1];
        bR.x = Br[(k0    ) * MSTR + bcol];
        bR.y = Br[(k0 + 1) * MSTR + bcol];
        bI.x = sB * Bi[(k0    ) * MSTR + bcol];
        bI.y = sB * Bi[(k0 + 1) * MSTR + bcol];
        v2f aIn = -aI;
        accR = wmma4(aR,  bR, accR);   // + Ar*Br
        accR = wmma4(aIn, bI, accR);   // - Ai*Bi
        accI = wmma4(aR,  bI, accI);   // + Ar*Bi
        accI = wmma4(aI,  bR, accI);   // + Ai*Br
      }
#pragma unroll
      for (int r = 0; r < 8; ++r) {
        const int drow = tm * 16 + r + hi * 8;
        Dr[drow * MSTR + tn * 16 + ml] = accR[r];
        Di[drow * MSTR + tn * 16 + ml] = accI[r];
      }
    }
  }
}

// ---------------------------------------------------------------------------
// 1024-pt complex FFT via radix-32x32 Cooley-Tukey, one wave.
//   n = 32*n1 + n2  ->  element n lives at LDS (n>>5)*MSTR + (n&31)
//   result X[k], k = k1 + 32*k2 -> lives at (k&31)*MSTR + (k>>5)
// conj = +1 forward (tables hold e^{-i...}), -1 inverse (unscaled).
// ---------------------------------------------------------------------------
__device__ __forceinline__ void fft1024(
    float* Xr, float* Xi, float* Tr, float* Ti,
    const float* Fr, const float* Fi,
    const float* Twr, const float* Twi, float conj, int lane) {
  // stage 1: T = F * X
  cmatmul32(Tr, Ti, Fr, Fi, conj, Xr, Xi, 1.0f, lane);
  __syncthreads();
  // twiddle: T[k1][n2] *= exp(-+ 2*pi*i*k1*n2/1024)
  for (int t = 0; t < 32; ++t) {
    const int idx = t * MSTR + lane;
    const float tr = Twr[idx], ti = conj * Twi[idx];
    const float xr = Tr[idx],  xi = Ti[idx];
    Tr[idx] = xr * tr - xi * ti;
    Ti[idx] = xr * ti + xi * tr;
  }
  __syncthreads();
  // stage 2: X = T * F   (DFT matrix is symmetric)
  cmatmul32(Xr, Xi, Tr, Ti, 1.0f, Fr, Fi, conj, lane);
  __syncthreads();
}

// ---------------------------------------------------------------------------
__device__ __forceinline__ float blockReduceSum(float v, float* red) {
  const int tid = threadIdx.x;
  red[tid] = v;
  __syncthreads();
  for (int s = blockDim.x >> 1; s > 0; s >>= 1) {
    if (tid < s) red[tid] += red[tid + s];
    __syncthreads();
  }
  const float r = red[0];
  __syncthreads();
  return r;
}

// ---------------------------------------------------------------------------
// K0: DFT-32 matrix + radix twiddle tables (resident in L2 afterwards)
// ---------------------------------------------------------------------------
__global__ void k_tables(float* Fr, float* Fi, float* Twr, float* Twi) {
  for (int idx = threadIdx.x; idx < 1024; idx += blockDim.x) {
    const int r = idx >> 5, c = idx & 31;
    float s1, c1, s2, c2;
    sincosf((float)((r * c) & 31) * (6.28318530717958647692f / 32.0f), &s1, &c1);
    sincosf((float)(r * c) * (6.28318530717958647692f / 1024.0f), &s2, &c2);
    Fr[r * MSTR + c] = c1;  Fi[r * MSTR + c] = -s1;
    Twr[r * MSTR + c] = c2; Twi[r * MSTR + c] = -s2;
  }
}

// K1: gaussian kernels g[i][j], j in [0,15): pdf/(sum+0.01)
__global__ void k_gauss(const float* sig, const float* cen, float* g) {
  const int i = threadIdx.x;
  if (i < KN) {
    const float s = sig[i] + 0.001f;
    const float c = cen[i];
    float pdf[KS];
    float sum = 0.f;
    for (int j = 0; j < KS; ++j) {
      const float t = ((float)j - 7.0f - c) / s;
      pdf[j] = expf(-0.5f * t * t);
      sum += pdf[j];
    }
    const float inv = 1.0f / (sum + 0.01f);
    for (int j = 0; j < KS; ++j) g[i * 16 + j] = pdf[j] * inv;
  }
}

// K2: channel sum of x, replicate-padded by K/2=7 each side
__global__ void k_chansum(const float* __restrict__ x, float* __restrict__ spad) {
  const int b = blockIdx.x;
  for (int l = threadIdx.x; l < Ll; l += blockDim.x) {
    float acc = 0.f;
    const float* xb = x + (size_t)b * Cc * Ll + l;
    for (int c = 0; c < Cc; ++c) acc += xb[(size_t)c * Ll];
    spad[b * SPSTR + 7 + l] = acc;
  }
  __syncthreads();
  if (threadIdx.x < 7) {
    spad[b * SPSTR + threadIdx.x]            = spad[b * SPSTR + 7];
    spad[b * SPSTR + 7 + Ll + threadIdx.x]   = spad[b * SPSTR + 7 + Ll - 1];
  }
}

// K3: blur(i,b,l) = sum_j spad[b][l+j] * g[i][j]  ('VALID' on padded row)
__global__ void k_blur(const float* __restrict__ spad, const float* __restrict__ g,
                       float* __restrict__ blur) {
  const int i = blockIdx.x / Bb, b = blockIdx.x % Bb;
  float gk[KS];
  for (int j = 0; j < KS; ++j) gk[j] = g[i * 16 + j];
  for (int l = threadIdx.x; l < Ll; l += blockDim.x) {
    const float* sp = spad + b * SPSTR + l;
    float acc = 0.f;
#pragma unroll
    for (int j = 0; j < KS; ++j) acc += sp[j] * gk[j];
    blur[((size_t)i * Bb + b) * Ll + l] = acc;
  }
}

// K4: forward FFT of q rows (16) and blur rows (128). One wave per row.
__global__ void k_fftrows(const float* __restrict__ q, const float* __restrict__ blur,
                          const float* Fr_g, const float* Fi_g,
                          const float* Twr_g, const float* Twi_g,
                          float* qfr, float* qfi, float* vfr, float* vfi) {
  __shared__ float Fr[MSZ], Fi[MSZ], Twr[MSZ], Twi[MSZ];
  __shared__ float Xr[MSZ], Xi[MSZ], Tr[MSZ], Ti[MSZ];
  const int lane = threadIdx.x;
  for (int idx = lane; idx < MSZ; idx += 32) {
    Fr[idx] = Fr_g[idx]; Fi[idx] = Fi_g[idx];
    Twr[idx] = Twr_g[idx]; Twi[idx] = Twi_g[idx];
  }
  const int rid = blockIdx.x;
  const float* src = (rid < Bb) ? (q + (size_t)rid * Ll)
                                : (blur + (size_t)(rid - Bb) * Ll);
  for (int t = 0; t < 32; ++t) {
    Xr[t * MSTR + lane] = src[t * 32 + lane];
    Xi[t * MSTR + lane] = 0.f;
  }
  __syncthreads();
  fft1024(Xr, Xi, Tr, Ti, Fr, Fi, Twr, Twi, 1.0f, lane);
  float* dr; float* di; size_t base;
  if (rid < Bb) { dr = qfr; di = qfi; base = (size_t)rid * Ll; }
  else          { dr = vfr; di = vfi; base = (size_t)(rid - Bb) * Ll; }
  for (int t = 0; t < 32; ++t) {
    const int k = t * 32 + lane;
    const int pidx = lane * MSTR + t;   // (k&31)*MSTR + (k>>5)
    dr[base + k] = Xr[pidx];
    di[base + k] = Xi[pidx];
  }
}

// K5: complex means mx[i] (over b,l; vfft is channel-constant) and my
__global__ void k_means(const float* qfr, const float* qfi,
                        const float* vfr, const float* vfi,
                        float* mx, float* my) {
  __shared__ float red[256];
  const int blk = blockIdx.x;
  const float* pr; const float* pi;
  if (blk < KN) { pr = vfr + (size_t)blk * Bb * Ll; pi = vfi + (size_t)blk * Bb * Ll; }
  else          { pr = qfr; pi = qfi; }
  const int n = Bb * Ll;
  float sr = 0.f, si = 0.f;
  for (int idx = threadIdx.x; idx < n; idx += blockDim.x) { sr += pr[idx]; si += pi[idx]; }
  sr = blockReduceSum(sr, red);
  si = blockReduceSum(si, red);
  if (threadIdx.x == 0) {
    const float inv = 1.0f / (float)n;
    if (blk < KN) { mx[blk * 2] = sr * inv; mx[blk * 2 + 1] = si * inv; }
    else          { my[0] = sr * inv;       my[1] = si * inv; }
  }
}

// K6: per (i,b): a = sum (v-mx)^2, bq = sum (q-mx)^2 (complex), then
//     w(l) = 1 - exp(-0.1 * |cov| / |a*bq + 0.001|)
__global__ void k_w(const float* qfr, const float* qfi,
                    const float* vfr, const float* vfi,
                    const float* mx, const float* my, float* wb) {
  __shared__ float red[256];
  const int i = blockIdx.x / Bb, b = blockIdx.x % Bb;
  const float mr = mx[i * 2], mi = mx[i * 2 + 1];
  const float myr = my[0], myi = my[1];
  const float* vr = vfr + ((size_t)i * Bb + b) * Ll;
  const float* vi = vfi + ((size_t)i * Bb + b) * Ll;
  const float* qr = qfr + (size_t)b * Ll;
  const float* qi = qfi + (size_t)b * Ll;
  float ar = 0.f, ai = 0.f, br = 0.f, bi = 0.f;
  for (int l = threadIdx.x; l < Ll; l += blockDim.x) {
    const float cr = vr[l] - mr, ci = vi[l] - mi;
    ar += cr * cr - ci * ci; ai += 2.f * cr * ci;
    const float dr = qr[l] - mr, di = qi[l] - mi;
    br += dr * dr - di * di; bi += 2.f * dr * di;
  }
  ar = blockReduceSum(ar, red);
  ai = blockReduceSum(ai, red);
  br = blockReduceSum(br, red);
  bi = blockReduceSum(bi, red);
  const float Dr = ar * br - ai * bi + 0.001f;
  const float Di = ar * bi + ai * br;
  const float invD = rsqrtf(Dr * Dr + Di * Di);   // 1/|a*b+0.001|
  float* wrow = wb + ((size_t)i * Bb + b) * Ll;
  for (int l = threadIdx.x; l < Ll; l += blockDim.x) {
    const float cr = vr[l] - mr,  ci = vi[l] - mi;
    const float er = qr[l] - myr, ei = qi[l] - myi;
    const float magcov = sqrtf((cr * cr + ci * ci) * (er * er + ei * ei));
    wrow[l] = 1.0f - expf(-0.1f * (magcov * invD));
  }
}

// K7 (main): per (b,c) wave — forward FFT of x row, then for each i:
//   vfilt = qfft*(1-w) + xfft*w ; o = |ifft(vfilt)| + x ; BN partials.
__global__ void __launch_bounds__(32)
k_main(const float* __restrict__ x, const float* __restrict__ qfr,
       const float* __restrict__ qfi, const float* __restrict__ wb,
       const float* Fr_g, const float* Fi_g, const float* Twr_g, const float* Twi_g,
       float* __restrict__ ob, float* __restrict__ ppart) {
  __shared__ float Fr[MSZ], Fi[MSZ], Twr[MSZ], Twi[MSZ];
  __shared__ float Xr[MSZ], Xi[MSZ], Tr[MSZ], Ti[MSZ];
  __shared__ float xfr[MSZ], xfi[MSZ];
  __shared__ float xrow[Ll], qlr[Ll], qli[Ll], sml[Ll], smq[Ll];
  const int lane = threadIdx.x;
  const int bc = blockIdx.x;
  const int b = bc / Cc;
  for (int idx = lane; idx < MSZ; idx += 32) {
    Fr[idx] = Fr_g[idx]; Fi[idx] = Fi_g[idx];
    Twr[idx] = Twr_g[idx]; Twi[idx] = Twi_g[idx];
  }
  const float* xp  = x   + (size_t)bc * Ll;
  const float* qrp = qfr + (size_t)b * Ll;
  const float* qip = qfi + (size_t)b * Ll;
  for (int t = 0; t < 32; ++t) {
    const int n = t * 32 + lane;
    const float xv = xp[n];
    xrow[n] = xv;
    Xr[t * MSTR + lane] = xv;
    Xi[t * MSTR + lane] = 0.f;
    qlr[n] = qrp[n]; qli[n] = qip[n];
    sml[n] = 0.f;    smq[n] = 0.f;
  }
  __syncthreads();
  fft1024(Xr, Xi, Tr, Ti, Fr, Fi, Twr, Twi, 1.0f, lane);   // xfft
  for (int idx = lane; idx < MSZ; idx += 32) { xfr[idx] = Xr[idx]; xfi[idx] = Xi[idx]; }
  __syncthreads();
  for (int i = 0; i < KN; ++i) {
    const float* wrow = wb + ((size_t)i * Bb + b) * Ll;
    __builtin_prefetch(wrow, 0, 1);                        // global_prefetch_b8
    for (int t = 0; t < 32; ++t) {
      const int k = t * 32 + lane;
      const float wv = wrow[k];
      const int pidx = lane * MSTR + t;                    // xfft[k] (perm layout)
      Xr[t * MSTR + lane] = qlr[k] * (1.f - wv) + xfr[pidx] * wv;
      Xi[t * MSTR + lane] = qli[k] * (1.f - wv) + xfi[pidx] * wv;
    }
    __syncthreads();
    fft1024(Xr, Xi, Tr, Ti, Fr, Fi, Twr, Twi, -1.0f, lane); // unscaled inverse
    float* orow = ob + ((size_t)bc * KN + i) * Ll;
    for (int t = 0; t < 32; ++t) {
      const int n = t * 32 + lane;
      const int pidx = lane * MSTR + t;
      const float yr = Xr[pidx] * (1.0f / 1024.0f);
      const float yi = Xi[pidx] * (1.0f / 1024.0f);
      const float ov = sqrtf(yr * yr + yi * yi) + xrow[n];
      orow[n] = ov;
      sml[n] += ov;
      smq[n] += ov * ov;
    }
    __syncthreads();
  }
  float* pp = ppart + (size_t)bc * 2048;
  for (int t = 0; t < 32; ++t) {
    const int n = t * 32 + lane;
    pp[n]        = sml[n];
    pp[1024 + n] = smq[n];
  }
}

// K8: finalize BN stats per l (coalesced column reduction over partials)
__global__ void k_bnstats(const float* __restrict__ ppart,
                          float* __restrict__ meanv, float* __restrict__ rstdv) {
  const int l = blockIdx.x * blockDim.x + threadIdx.x;
  if (l >= Ll) return;
  float s = 0.f, sq = 0.f;
  for (int pb = 0; pb < Bb * Cc; ++pb) {
    const float* pp = ppart + (size_t)pb * 2048;
    s  += pp[l];
    sq += pp[1024 + l];
  }
  const float inv = 1.0f / (float)(Bb * Cc * KN);
  const float m = s * inv;
  const float var = sq * inv - m * m;
  meanv[l] = m;
  rstdv[l] = rsqrtf(var + 1e-5f);
}

// K9: normalize o (BN affine), scores = q . o_hat, softmax over KNUM, pool.
__global__ void k_attn(const float* __restrict__ ob, const float* __restrict__ qv,
                       const float* __restrict__ meanv, const float* __restrict__ rstdv,
                       const float* __restrict__ gamma, const float* __restrict__ beta,
                       float* __restrict__ out) {
  __shared__ float oh[KN * Ll];   // 32 KB — all 8 normalized rows of this (b,c)
  __shared__ float qs[Ll];
  __shared__ float red[256];
  __shared__ float scs[KN];
  const int bc = blockIdx.x;
  const int b = bc / Cc;
  const int tid = threadIdx.x;
  for (int l = tid; l < Ll; l += blockDim.x) qs[l] = qv[(size_t)b * Ll + l];
  for (int i = 0; i < KN; ++i) {
    const float* orow = ob + ((size_t)bc * KN + i) * Ll;
    for (int l = tid; l < Ll; l += blockDim.x)
      oh[i * Ll + l] = (orow[l] - meanv[l]) * rstdv[l] * gamma[l] + beta[l];
  }
  __syncthreads();
  for (int i = 0; i < KN; ++i) {
    float s = 0.f;
    for (int l = tid; l < Ll; l += blockDim.x) s += qs[l] * oh[i * Ll + l];
    s = blockReduceSum(s, red);
    if (tid == 0) scs[i] = s;
    __syncthreads();
  }
  float mxv = scs[0];
  for (int i = 1; i < KN; ++i) mxv = fmaxf(mxv, scs[i]);
  float att[KN];
  float se = 0.f;
  for (int i = 0; i < KN; ++i) { att[i] = expf(scs[i] - mxv); se += att[i]; }
  const float inv = 1.0f / se;
  for (int i = 0; i < KN; ++i) att[i] *= inv;
  for (int l = tid; l < Ll; l += blockDim.x) {
    float acc = 0.f;
#pragma unroll
    for (int i = 0; i < KN; ++i) acc += att[i] * oh[i * Ll + l];
    out[(size_t)bc * Ll + l] = acc;
  }
}

// ---------------------------------------------------------------------------
extern "C" void kernel_launch(void* const* d_in, const int* in_sizes, int n_in,
                              void* d_out, int out_size, void* d_ws, size_t ws_size,
                              hipStream_t stream) {
  (void)in_sizes; (void)n_in; (void)out_size; (void)ws_size;
  const float* x     = (const float*)d_in[0];
  const float* q     = (const float*)d_in[1];
  const float* sig   = (const float*)d_in[2];
  const float* cen   = (const float*)d_in[3];
  const float* gamma = (const float*)d_in[4];
  const float* beta  = (const float*)d_in[5];
  float* out = (float*)d_out;

  float* W = (float*)d_ws;
  size_t p = 0;
  auto alloc = [&](size_t n) { size_t r = p; p += n; return r; };
  float* g     = W + alloc(KN * 16);
  float* spad  = W + alloc((size_t)Bb * SPSTR);
  float* qfr   = W + alloc((size_t)Bb * Ll);
  float* qfi   = W + alloc((size_t)Bb * Ll);
  float* vfr   = W + alloc((size_t)KN * Bb * Ll);
  float* vfi   = W + alloc((size_t)KN * Bb * Ll);
  float* blur  = W + alloc((size_t)KN * Bb * Ll);
  float* wb    = W + alloc((size_t)KN * Bb * Ll);
  float* mx    = W + alloc(KN * 2);
  float* my    = W + alloc(2);
  float* Fr    = W + alloc(MSZ);
  float* Fi    = W + alloc(MSZ);
  float* Twr   = W + alloc(MSZ);
  float* Twi   = W + alloc(MSZ);
  float* meanv = W + alloc(Ll);
  float* rstdv = W + alloc(Ll);
  float* ob    = W + alloc((size_t)Bb * Cc * KN * Ll);   // 268 MB
  float* ppart = W + alloc((size_t)Bb * Cc * 2048);      //  67 MB

  k_tables <<<1, 256, 0, stream>>>(Fr, Fi, Twr, Twi);
  k_gauss  <<<1, 32, 0, stream>>>(sig, cen, g);
  k_chansum<<<Bb, 256, 0, stream>>>(x, spad);
  k_blur   <<<KN * Bb, 256, 0, stream>>>(spad, g, blur);
  k_fftrows<<<Bb + KN * Bb, 32, 0, stream>>>(q, blur, Fr, Fi, Twr, Twi,
                                             qfr, qfi, vfr, vfi);
  k_means  <<<KN + 1, 256, 0, stream>>>(qfr, qfi, vfr, vfi, mx, my);
  k_w      <<<KN * Bb, 256, 0, stream>>>(qfr, qfi, vfr, vfi, mx, my, wb);
  k_main   <<<Bb * Cc, 32, 0, stream>>>(x, qfr, qfi, wb, Fr, Fi, Twr, Twi,
                                        ob, ppart);
  k_bnstats<<<(Ll + 255) / 256, 256, 0, stream>>>(ppart, meanv, rstdv);
  k_attn   <<<Bb * Cc, 256, 0, stream>>>(ob, q, meanv, rstdv, gamma, beta, out);
}